// ScorePredictor_81097572483639
// MI455X (gfx1250) — compile-verified
//
#include <hip/hip_runtime.h>
#include <hip/hip_bf16.h>

typedef float v2f __attribute__((ext_vector_type(2)));
typedef float v8f __attribute__((ext_vector_type(8)));

// One wave (32 lanes) scores 16 edges per group using V_WMMA_F32_16X16X4_F32.
// A-matrix: 16 gathered src rows, B-matrix: 16 gathered dst rows (as columns).
// Accumulate over K=128 in 32 steps of K=4; diagonal of the 16x16 result is
// the 16 dot products. Loads are exactly 2 float2 per lane per step -> zero
// redundant memory traffic vs. a scalar dot-product kernel.
__global__ void ScorePredictor_81097572483639_kernel(
    const float* __restrict__ h_drug,
    const float* __restrict__ d_disease,
    const int* __restrict__ src_idx,
    const int* __restrict__ dst_idx,
    float* __restrict__ out,
    int n_edges, int n_groups) {
  const int lane          = threadIdx.x & 31;
  const int waveInBlock   = threadIdx.x >> 5;
  const int wavesPerBlock = blockDim.x >> 5;
  const int groupStride   = gridDim.x * wavesPerBlock;
  int group = blockIdx.x * wavesPerBlock + waveInBlock;

  const int m  = lane & 15;   // A-row / B-col handled by this lane
  const int kh = lane >> 4;   // K-half: lanes 0-15 -> K{0,1}, lanes 16-31 -> K{2,3}

  for (; group < n_groups; group += groupStride) {
    const int base = group * 16;
    int e = base + m;
    if (e >= n_edges) e = 0;      // clamp tail; EXEC must stay all-ones for WMMA

    const int si = src_idx[e];
    const int di = dst_idx[e];
    const float* pa = h_drug    + (size_t)si * 128 + (kh << 1);
    const float* pb = d_disease + (size_t)di * 128 + (kh << 1);

    v8f c = {0.f, 0.f, 0.f, 0.f, 0.f, 0.f, 0.f, 0.f};
#pragma unroll
    for (int k0 = 0; k0 < 128; k0 += 4) {
      v2f a = *(const v2f*)(pa + k0);
      v2f b = *(const v2f*)(pb + k0);
      // 8 args: (neg_a, A, neg_b, B, c_mod, C, reuse_a, reuse_b)
      c = __builtin_amdgcn_wmma_f32_16x16x4_f32(false, a, false, b,
                                                (short)0, c, false, false);
    }

    // Diagonal extraction (32-bit C/D layout, 16x16):
    //   element (m,m): m<8 -> c[m]   held in lane m
    //                  m>=8 -> c[m-8] held in lane m+16
    const int srcLane = (m < 8) ? m : (m + 16);
    const int sel     = m & 7;
    float diag = 0.f;
#pragma unroll
    for (int i = 0; i < 8; ++i) {
      float t = __shfl(c[i], srcLane, 32);
      if (sel == i) diag = t;
    }

    const int eo = base + lane;
    if (lane < 16 && eo < n_edges) {
      float mean = diag * (1.0f / 128.0f);
      out[eo] = 1.0f / (1.0f + __expf(-mean));
    }
  }
}

extern "C" void kernel_launch(void* const* d_in, const int* in_sizes, int n_in,
                              void* d_out, int out_size, void* d_ws, size_t ws_size,
                              hipStream_t stream) {
  const float* h_drug    = (const float*)d_in[0];
  const float* d_disease = (const float*)d_in[1];
  const int*   src_idx   = (const int*)d_in[2];
  const int*   dst_idx   = (const int*)d_in[3];
  float*       out       = (float*)d_out;

  const int n_edges  = in_sizes[2];
  const int n_groups = (n_edges + 15) / 16;

  const int threads        = 256;             // 8 wave32 waves per block
  const int wavesPerBlock  = threads / 32;
  int blocks = (n_groups + wavesPerBlock - 1) / wavesPerBlock;
  if (blocks < 1) blocks = 1;

  ScorePredictor_81097572483639_kernel<<<blocks, threads, 0, stream>>>(
      h_drug, d_disease, src_idx, dst_idx, out, n_edges, n_groups);
}